// NeuralFieldSet_18605798326295
// MI455X (gfx1250) — compile-verified
//
#include <hip/hip_runtime.h>

#define NQ   32768
#define NF   1024
#define NH   32
#define KNN  8

typedef __attribute__((ext_vector_type(2))) float v2f;
typedef __attribute__((ext_vector_type(8))) float v8f;
typedef unsigned int u32;
typedef __attribute__((ext_vector_type(4))) u32 v4u;
typedef __attribute__((ext_vector_type(8))) int  v8i;
typedef __attribute__((ext_vector_type(4))) int  v4i;

// V_WMMA_F32_16X16X4_F32: D(16x16 f32) = A(16x4 f32) * B(4x16 f32) + C
__device__ __forceinline__ v8f wmma4(v2f a, v2f b, v8f c) {
    return __builtin_amdgcn_wmma_f32_16x16x4_f32(
        false, a, false, b, (short)0, c, false, false);
}

// ---------------------------------------------------------------------------
// Kernel 1: per-point kNN over all fields (field centers staged in LDS),
// local-coordinate transform (inverse quaternion rotate), softmax weights.
// Weights are zeroed when outside FIELD_RADIUS so the final reduction
// produces OUTSIDE (=0) automatically.
// ---------------------------------------------------------------------------
__global__ __launch_bounds__(256) void knn_kernel(
    const float* __restrict__ qp, const float* __restrict__ fp,
    const float* __restrict__ fo,
    int* __restrict__ counts, int* __restrict__ knn_f,
    float4* __restrict__ recPre)
{
    __shared__ float lfp[NF][4];
    const int tid = threadIdx.x;
    for (int i = tid; i < NF; i += 256) {
        lfp[i][0] = fp[i*3+0];
        lfp[i][1] = fp[i*3+1];
        lfp[i][2] = fp[i*3+2];
        lfp[i][3] = 0.0f;
    }
    __syncthreads();

    const int n = blockIdx.x * 256 + tid;
    const float qx = qp[n*3+0], qy = qp[n*3+1], qz = qp[n*3+2];

    float bd[KNN]; int bi[KNN];
    #pragma unroll
    for (int k = 0; k < KNN; ++k) { bd[k] = 1e30f; bi[k] = 0; }

    for (int j = 0; j < NF; ++j) {
        float dx = qx - lfp[j][0];
        float dy = qy - lfp[j][1];
        float dz = qz - lfp[j][2];
        float d2 = dx*dx + dy*dy + dz*dz;
        if (d2 < bd[KNN-1]) {
            bd[KNN-1] = d2; bi[KNN-1] = j;
            #pragma unroll
            for (int s = KNN-1; s > 0; --s) {
                if (bd[s] < bd[s-1]) {
                    float td = bd[s]; bd[s] = bd[s-1]; bd[s-1] = td;
                    int   ti = bi[s]; bi[s] = bi[s-1]; bi[s-1] = ti;
                }
            }
        }
    }

    float d[KNN];
    #pragma unroll
    for (int k = 0; k < KNN; ++k) d[k] = sqrtf(bd[k]);
    const bool inside = d[0] < 0.5f;     // FIELD_RADIUS

    float e[KNN]; float esum = 0.0f;
    #pragma unroll
    for (int k = 0; k < KNN; ++k) { e[k] = __expf(-10.0f * (d[k] - d[0])); esum += e[k]; }
    const float escale = inside ? (1.0f / esum) : 0.0f;

    #pragma unroll
    for (int k = 0; k < KNN; ++k) {
        const int f = bi[k];
        atomicAdd(&counts[f], 1);
        const float4 q4 = ((const float4*)fo)[f];   // (w,x,y,z) real-first
        float vx = qx - lfp[f][0];
        float vy = qy - lfp[f][1];
        float vz = qz - lfp[f][2];
        // rotate v by conj(q): u = -q.xyz, w = q.w ; v' = v + w*t + u x t, t = 2(u x v)
        const float uw = q4.x, ux = -q4.y, uy = -q4.z, uz = -q4.w;
        const float tx = 2.0f * (uy*vz - uz*vy);
        const float ty = 2.0f * (uz*vx - ux*vz);
        const float tz = 2.0f * (ux*vy - uy*vx);
        float lx = vx + uw*tx + (uy*tz - uz*ty);
        float ly = vy + uw*ty + (uz*tx - ux*tz);
        float lz = vz + uw*tz + (ux*ty - uy*tx);
        lx *= 2.0f; ly *= 2.0f; lz *= 2.0f;     // / FIELD_RADIUS
        knn_f[n*KNN + k]  = f;
        recPre[n*KNN + k] = make_float4(lx, ly, lz, e[k] * escale);
    }
}

// ---------------------------------------------------------------------------
// Kernel 2: exclusive prefix sum over per-field counts -> CSR offsets.
// ---------------------------------------------------------------------------
__global__ __launch_bounds__(1024) void scan_kernel(
    const int* __restrict__ counts, int* __restrict__ offsets, int* __restrict__ cursor)
{
    __shared__ int s[NF];
    const int t = threadIdx.x;
    const int c = counts[t];
    s[t] = c;
    __syncthreads();
    for (int off = 1; off < NF; off <<= 1) {
        int v = (t >= off) ? s[t - off] : 0;
        __syncthreads();
        s[t] += v;
        __syncthreads();
    }
    offsets[t] = s[t] - c;
    if (t == NF - 1) offsets[NF] = s[t];
    cursor[t] = 0;
}

// ---------------------------------------------------------------------------
// Kernel 3: scatter (n,k) records into field-grouped CSR order.
// ---------------------------------------------------------------------------
__global__ __launch_bounds__(256) void fill_kernel(
    const int* __restrict__ knn_f, const float4* __restrict__ recPre,
    const int* __restrict__ offsets, int* __restrict__ cursor,
    float4* __restrict__ recA, int* __restrict__ recN)
{
    const int nk = blockIdx.x * 256 + threadIdx.x;
    const int f = knn_f[nk];
    const int slot = offsets[f] + atomicAdd(&cursor[f], 1);
    recA[slot] = recPre[nk];
    recN[slot] = nk;
}

// ---------------------------------------------------------------------------
// Kernel 4: per-field batched MLP via fp32 WMMA (16x16x4).
// One block per field, 4 waves. W2 (4KB) is DMA'd into LDS by the Tensor
// Data Mover with pad_enable producing the pitch-33 (bank-conflict-free)
// layout directly. All WMMA B-fragments are hoisted into registers; LDS in
// the tile loop is only used for the D-layout -> A-layout activation
// transpose.
// ---------------------------------------------------------------------------
__global__ __launch_bounds__(128) void mlp_kernel(
    const float* __restrict__ W1, const float* __restrict__ B1,
    const float* __restrict__ W2, const float* __restrict__ B2,
    const float* __restrict__ W3, const float* __restrict__ B3,
    const int* __restrict__ offsets,
    const float4* __restrict__ recA, const int* __restrict__ recN,
    float* __restrict__ outNK)
{
    const int f   = blockIdx.x;
    const int tid = threadIdx.x;
    const int start = offsets[f];
    const int cnt   = offsets[f + 1] - start;
    if (cnt == 0) return;

    __shared__ float sW1[NH][4];       // [n][k], k=3 holds b1 (bias folded, K=4)
    __shared__ float sW2[NH][33];      // pitch 33: written by TDM with padding
    __shared__ float sW3[16][33];      // rows 4..15 zero (N padded to 16)
    __shared__ float sB2[NH];
    __shared__ float sB3[16];
    __shared__ float sH [4][16][33];   // per-wave activation staging [M][N]
    __shared__ float sMW[4][16];       // per-wave tile weights
    __shared__ int   sMN[4][16];       // per-wave tile nk indices (-1 = pad)

    // --- Tensor Data Mover: DMA W2[f] (1024 f32) -> sW2 with +1 DWORD pad
    //     every 32 DWORDs (pad_interval=4 -> 32 DWORDs, pad_amount=0 -> 1).
    if (tid == 0) {
        const uint64_t ga  = (uint64_t)(const void*)(W2 + (size_t)f * NH * NH);
        const u32      lds = (u32)(uintptr_t)&sW2[0][0];
        v4u g0 = {0u, 0u, 0u, 0u};
        g0[0] = 1u;                                        // count=1 (valid)
        g0[1] = lds;                                       // lds_addr
        g0[2] = (u32)ga;                                   // global_addr[31:0]
        g0[3] = (u32)((ga >> 32) & 0x01FFFFFFu) | (2u << 30); // addr[56:32] | type=2
        v8i g1 = {0, 0, 0, 0, 0, 0, 0, 0};
        g1[0] = (2 << 16) | (1 << 20) | (4 << 22);         // data_size=4B, pad_en, pad_int=32dw
        g1[1] = (int)((u32)(NH * NH) << 16);               // tensor_dim0 = 1024 (bits 63:48)
        g1[2] = (1 << 16);                                 // tensor_dim1 = 1   (bits 95:80)
        g1[3] = (int)((u32)(NH * NH) << 16);               // tile_dim0 = 1024  (bits 127:112)
        g1[4] = 1;                                         // tile_dim1 = 1     (bits 143:128)
        g1[5] = NH * NH;                                   // tensor_dim0_stride (bits 191:160)
        v4i gz = {0, 0, 0, 0};
        v8i gz8 = {0, 0, 0, 0, 0, 0, 0, 0};
        __builtin_amdgcn_tensor_load_to_lds(g0, g1, gz, gz, gz8, 0);
    }

    // --- stage the small weight arrays with regular loads (overlaps TDM) ---
    for (int i = tid; i < NH*3; i += 128) sW1[i/3][i%3] = W1[f*NH*3 + i];
    for (int i = tid; i < NH;   i += 128) sW1[i][3]     = B1[f*NH + i];
    for (int i = tid; i < 16*NH;i += 128) {
        const int row = i / NH, col = i % NH;
        sW3[row][col] = (row < 4) ? W3[f*4*NH + row*NH + col] : 0.0f;
    }
    for (int i = tid; i < NH; i += 128) sB2[i] = B2[f*NH + i];
    for (int i = tid; i < 16; i += 128) sB3[i] = (i < 4) ? B3[f*4 + i] : 0.0f;

    __builtin_amdgcn_s_wait_tensorcnt(0);   // wave0: TDM done; others: no-op
    __syncthreads();

    const int wv   = tid >> 5;
    const int lane = tid & 31;
    const int m    = lane & 15;    // record row (A) / output column n (B,C,D)
    const int kh   = lane >> 4;    // K-half for A/B layouts
    const int ntiles = (cnt + 15) >> 4;

    // --- hoist all loop-invariant B-fragments / biases into registers ---
    v2f bw1[2]; v2f bw2[8][2]; v2f bw3[8];
    float cb2[2];
    #pragma unroll
    for (int t = 0; t < 2; ++t) {
        const int nn = m + 16*t;
        bw1[t][0] = sW1[nn][2*kh + 0];
        bw1[t][1] = sW1[nn][2*kh + 1];
        cb2[t]    = sB2[nn];
        #pragma unroll
        for (int c = 0; c < 8; ++c) {
            bw2[c][t][0] = sW2[nn][4*c + 2*kh + 0];
            bw2[c][t][1] = sW2[nn][4*c + 2*kh + 1];
        }
    }
    #pragma unroll
    for (int c = 0; c < 8; ++c) {
        bw3[c][0] = sW3[m][4*c + 2*kh + 0];
        bw3[c][1] = sW3[m][4*c + 2*kh + 1];
    }
    const float cb3 = sB3[m];

    for (int tile = wv; tile < ntiles; tile += 4) {
        const int r = start + tile*16 + m;
        float4 ra = make_float4(0.f, 0.f, 0.f, 0.f);
        int nk = -1;
        if (r < start + cnt) { ra = recA[r]; nk = recN[r]; }
        if (kh == 0) { sMW[wv][m] = ra.w; sMN[wv][m] = nk; }

        // ---- layer 1: h1 = relu([local,1](16x4) @ W1ext^T(4x32)) ----
        v2f a;                                  // A 16x4: a[j] = ext[m][j+2*kh]
        a[0] = kh ? ra.z : ra.x;
        a[1] = kh ? 1.0f : ra.y;
        v8f h[2];
        #pragma unroll
        for (int t = 0; t < 2; ++t) {
            v8f z = {0.f,0.f,0.f,0.f,0.f,0.f,0.f,0.f};
            h[t] = wmma4(a, bw1[t], z);
        }
        #pragma unroll
        for (int t = 0; t < 2; ++t) {
            const int nn = m + 16*t;
            #pragma unroll
            for (int v = 0; v < 8; ++v)
                sH[wv][v + 8*kh][nn] = fmaxf(h[t][v], 0.0f);
        }

        // ---- layer 2: h2 = relu(h1(16x32) @ W2^T(32x32) + b2), K chained ----
        v8f acc2[2];
        #pragma unroll
        for (int t = 0; t < 2; ++t) {
            #pragma unroll
            for (int v = 0; v < 8; ++v) acc2[t][v] = cb2[t];
        }
        #pragma unroll
        for (int c = 0; c < 8; ++c) {
            v2f a2;
            a2[0] = sH[wv][m][4*c + 2*kh + 0];
            a2[1] = sH[wv][m][4*c + 2*kh + 1];
            #pragma unroll
            for (int t = 0; t < 2; ++t)
                acc2[t] = wmma4(a2, bw2[c][t], acc2[t]);
        }
        #pragma unroll
        for (int t = 0; t < 2; ++t) {
            const int nn = m + 16*t;
            #pragma unroll
            for (int v = 0; v < 8; ++v)
                sH[wv][v + 8*kh][nn] = fmaxf(acc2[t][v], 0.0f);
        }

        // ---- layer 3: out = h2(16x32) @ W3ext^T(32x16pad) + b3 ----
        v8f acc3;
        #pragma unroll
        for (int v = 0; v < 8; ++v) acc3[v] = cb3;
        #pragma unroll
        for (int c = 0; c < 8; ++c) {
            v2f a3;
            a3[0] = sH[wv][m][4*c + 2*kh + 0];
            a3[1] = sH[wv][m][4*c + 2*kh + 1];
            acc3 = wmma4(a3, bw3[c], acc3);
        }

        // ---- scale by softmax weight and scatter to per-(n,k) slot ----
        if (m < 4) {
            #pragma unroll
            for (int v = 0; v < 8; ++v) {
                const int M = v + 8*kh;
                const int onk = sMN[wv][M];
                if (onk >= 0) outNK[onk*4 + m] = sMW[wv][M] * acc3[v];
            }
        }
    }
}

// ---------------------------------------------------------------------------
// Kernel 5: deterministic fixed-order combine over K (float4 vectorized).
// ---------------------------------------------------------------------------
__global__ __launch_bounds__(256) void reduce_kernel(
    const float4* __restrict__ outNK4, float4* __restrict__ out4)
{
    const int n = blockIdx.x * 256 + threadIdx.x;   // [0, NQ)
    float sx = 0.f, sy = 0.f, sz = 0.f, sw = 0.f;
    #pragma unroll
    for (int k = 0; k < KNN; ++k) {
        const float4 v = outNK4[n*KNN + k];
        sx += v.x; sy += v.y; sz += v.z; sw += v.w;
    }
    out4[n] = make_float4(sx, sy, sz, sw);
}

extern "C" void kernel_launch(void* const* d_in, const int* in_sizes, int n_in,
                              void* d_out, int out_size, void* d_ws, size_t ws_size,
                              hipStream_t stream) {
    const float* qp = (const float*)d_in[0];
    const float* fp = (const float*)d_in[1];
    const float* fo = (const float*)d_in[2];
    const float* W1 = (const float*)d_in[3];
    const float* B1 = (const float*)d_in[4];
    const float* W2 = (const float*)d_in[5];
    const float* B2 = (const float*)d_in[6];
    const float* W3 = (const float*)d_in[7];
    const float* B3 = (const float*)d_in[8];

    char* w = (char*)d_ws;
    auto alloc = [&](size_t bytes) -> void* {
        void* p = (void*)w;
        w += (bytes + 255) & ~(size_t)255;
        return p;
    };
    int*    counts  = (int*)   alloc(sizeof(int)    * NF);
    int*    offsets = (int*)   alloc(sizeof(int)    * (NF + 1));
    int*    cursor  = (int*)   alloc(sizeof(int)    * NF);
    int*    knn_f   = (int*)   alloc(sizeof(int)    * NQ * KNN);
    float4* recPre  = (float4*)alloc(sizeof(float4) * NQ * KNN);
    float4* recA    = (float4*)alloc(sizeof(float4) * NQ * KNN);
    int*    recN    = (int*)   alloc(sizeof(int)    * NQ * KNN);
    float*  outNK   = (float*) alloc(sizeof(float)  * NQ * KNN * 4);

    (void)hipMemsetAsync(counts, 0, sizeof(int) * NF, stream);
    knn_kernel  <<<NQ/256, 256, 0, stream>>>(qp, fp, fo, counts, knn_f, recPre);
    scan_kernel <<<1, NF, 0, stream>>>(counts, offsets, cursor);
    fill_kernel <<<(NQ*KNN)/256, 256, 0, stream>>>(knn_f, recPre, offsets, cursor, recA, recN);
    mlp_kernel  <<<NF, 128, 0, stream>>>(W1, B1, W2, B2, W3, B3, offsets, recA, recN, outNK);
    reduce_kernel<<<NQ/256, 256, 0, stream>>>((const float4*)outNK, (float4*)d_out);
}